// GraphEdgeVertLayer_78408922956199
// MI455X (gfx1250) — compile-verified
//
#include <hip/hip_runtime.h>

// ---------------------------------------------------------------------------
// GraphEdgeVertLayer for MI455X (gfx1250), f32 WMMA path.
//   B=32, N=128, F=128 everywhere.
// Roofline: 256MB e_in read + 256MB e_out write ~= 22us @ 23.3 TB/s.
// 17.6 GFLOP f32 via V_WMMA_F32_16X16X4_F32 (exact f32, matches reference).
// ---------------------------------------------------------------------------

typedef __attribute__((ext_vector_type(2))) float v2f;
typedef __attribute__((ext_vector_type(8))) float v8f;

#define Bb   32
#define Nn   128
#define Ff   128
#define ROWS (Bb * Nn)          // 4096 vertex rows

__device__ __forceinline__ v8f wmma_f32_k4(v2f a, v2f b, v8f c) {
  // D(16x16,f32) = A(16x4,f32) * B(4x16,f32) + C
  return __builtin_amdgcn_wmma_f32_16x16x4_f32(
      /*neg_a=*/false, a, /*neg_b=*/false, b,
      /*c_mod=*/(short)0, c, /*reuse_a=*/false, /*reuse_b=*/false);
}

// ---------------------------------------------------------------------------
// Stage 1: e_v[b,n,:] = v_in[b,n,:] @ W_ev + b_ev      (4096 x 128, K=128)
// 1 wave = one 16-row strip x full 128 cols (8 col-tiles).
// ---------------------------------------------------------------------------
__global__ __launch_bounds__(128) void ev_kernel(
    const float* __restrict__ v_in, const float* __restrict__ W_ev,
    const float* __restrict__ b_ev, float* __restrict__ e_v) {
  const int lane = threadIdx.x & 31;
  const int wave = threadIdx.x >> 5;
  const int half = lane >> 4;          // 0: K pair (k,k+1), 1: (k+2,k+3)
  const int l15  = lane & 15;
  const int strip = blockIdx.x * 4 + wave;   // 0..255
  const int row0  = strip * 16;

  const float* __restrict__ arow = v_in + (size_t)(row0 + l15) * Ff;

  v8f acc[8] = {};
  for (int k = 0; k < Ff; k += 4) {
    const int kb = k + 2 * half;
    v2f a = *(const v2f*)(arow + kb);
    const float* __restrict__ w0 = W_ev + (size_t)kb * Ff;
#pragma unroll
    for (int t = 0; t < 8; ++t) {
      const int g = t * 16 + l15;
      v2f bf;
      bf.x = w0[g];
      bf.y = w0[Ff + g];
      acc[t] = wmma_f32_k4(a, bf, acc[t]);
    }
  }
#pragma unroll
  for (int t = 0; t < 8; ++t) {
    const int g = t * 16 + l15;
    const float bias = b_ev[g];
#pragma unroll
    for (int r = 0; r < 8; ++r) {
      const int m = r + half * 8;                       // D: M=r (lanes 0-15), M=r+8 (16-31)
      e_v[(size_t)(row0 + m) * Ff + g] = acc[t][r] + bias;
    }
  }
}

// ---------------------------------------------------------------------------
// Stage 2: e_out[b,m,n,:] = relu(e_in[b,m,n,:] @ W_e + b_e + e_v[b,n,:] + e_v[b,m,:])
//          partial[b,mt,n,:] = sum over this block's 16 m of e_out
// Grid (nt=8, mt=8, b=32), 256 threads (8 waves). Wave w does strips 2w,2w+1
// (strip s => m = m0+s, A-rows are the 16 n of this n-tile).
// ---------------------------------------------------------------------------
__global__ __launch_bounds__(256) void edge_kernel(
    const float* __restrict__ e_in, const float* __restrict__ W_e,
    const float* __restrict__ b_e, const float* __restrict__ e_v,
    float* __restrict__ e_out, float* __restrict__ partial) {
  __shared__ float sEvN[16 * Ff];   // e_v rows n0..n0+15
  __shared__ float sEvM[16 * Ff];   // e_v rows m0..m0+15
  __shared__ float sRed[16 * Ff];   // deterministic m-reduction buffer
  __shared__ float sBe[Ff];

  const int nt = blockIdx.x, mt = blockIdx.y, b = blockIdx.z;
  const int n0 = nt * 16, m0 = mt * 16;
  const int tid  = threadIdx.x;
  const int lane = tid & 31;
  const int wave = tid >> 5;
  const int half = lane >> 4;
  const int l15  = lane & 15;

  { // cooperative staging (float4 = b128 vector loads)
    const float4* srcN = (const float4*)(e_v + ((size_t)b * Nn + n0) * Ff);
    const float4* srcM = (const float4*)(e_v + ((size_t)b * Nn + m0) * Ff);
    float4* dN = (float4*)sEvN;
    float4* dM = (float4*)sEvM;
    float4* dR = (float4*)sRed;
    const float4 z4 = {0.f, 0.f, 0.f, 0.f};
    for (int i = tid; i < 512; i += 256) { dN[i] = srcN[i]; dM[i] = srcM[i]; dR[i] = z4; }
    if (tid < Ff) sBe[tid] = b_e[tid];
  }
  __syncthreads();

  v8f psum[8] = {};   // per-wave m-partials: one element per (n_local, g) pair

  for (int s2 = 0; s2 < 2; ++s2) {
    const int s = wave * 2 + s2;          // strip = local m index, 0..15
    const int m = m0 + s;
    // A row j (=n_local) of this strip: e_in[b, m, n0+j, :]
    const float* __restrict__ arow =
        e_in + (((size_t)b * Nn + m) * Nn + n0 + l15) * Ff;

    v8f acc[8] = {};
    for (int k = 0; k < Ff; k += 4) {
      const int kb = k + 2 * half;
      v2f a = *(const v2f*)(arow + kb);
      const float* __restrict__ w0 = W_e + (size_t)kb * Ff;   // hot in WGP$/L2
#pragma unroll
      for (int t = 0; t < 8; ++t) {
        const int g = t * 16 + l15;
        v2f bf;
        bf.x = w0[g];
        bf.y = w0[Ff + g];
        acc[t] = wmma_f32_k4(a, bf, acc[t]);
      }
    }

    // fused epilogue: + b_e + e_v[n] + e_v[m], relu, store, accumulate m-sum
#pragma unroll
    for (int t = 0; t < 8; ++t) {
      const int g = t * 16 + l15;
      const float addc = sBe[g] + sEvM[s * Ff + g];
#pragma unroll
      for (int r = 0; r < 8; ++r) {
        const int nl = r + half * 8;
        float val = acc[t][r] + addc + sEvN[nl * Ff + g];
        val = fmaxf(val, 0.0f);
        e_out[(((size_t)b * Nn + m) * Nn + (n0 + nl)) * Ff + g] = val;
        psum[t][r] += val;
      }
    }
  }

  // Deterministic cross-wave reduction: waves add into sRed in fixed order.
  for (int w = 0; w < 8; ++w) {
    if (wave == w) {
#pragma unroll
      for (int t = 0; t < 8; ++t) {
        const int g = t * 16 + l15;
#pragma unroll
        for (int r = 0; r < 8; ++r) {
          const int nl = r + half * 8;
          sRed[nl * Ff + g] += psum[t][r];
        }
      }
    }
    __syncthreads();
  }

  // Unique writer per (b, mt, n-slice): plain stores, no atomics.
  for (int i = tid; i < 16 * Ff; i += 256) {
    const int nl = i >> 7, g = i & (Ff - 1);
    partial[(((size_t)b * 8 + mt) * Nn + (n0 + nl)) * Ff + g] = sRed[i];
  }
}

// ---------------------------------------------------------------------------
// Stage 3: per_v_e[b,n,:] = sum_mt partial[b,mt,n,:]   (deterministic)
// ---------------------------------------------------------------------------
__global__ __launch_bounds__(256) void reduce_kernel(
    const float* __restrict__ partial, float* __restrict__ per_v_e) {
  const int i = blockIdx.x * 256 + threadIdx.x;   // over 4096*128
  if (i < ROWS * Ff) {
    const int g = i & (Ff - 1);
    const int bn = i >> 7;
    const int n = bn & (Nn - 1);
    const int b = bn >> 7;
    float s = 0.0f;
#pragma unroll
    for (int mt = 0; mt < 8; ++mt)
      s += partial[(((size_t)b * 8 + mt) * Nn + n) * Ff + g];
    per_v_e[i] = s;
  }
}

// ---------------------------------------------------------------------------
// Stage 4: v_out = relu(concat(per_v_e, v_in) @ W_v + b_v)   (4096x128, K=256)
// ---------------------------------------------------------------------------
__global__ __launch_bounds__(128) void vert_kernel(
    const float* __restrict__ per_v_e, const float* __restrict__ v_in,
    const float* __restrict__ W_v, const float* __restrict__ b_v,
    float* __restrict__ v_out) {
  const int lane = threadIdx.x & 31;
  const int wave = threadIdx.x >> 5;
  const int half = lane >> 4;
  const int l15  = lane & 15;
  const int strip = blockIdx.x * 4 + wave;
  const int row0  = strip * 16;

  const float* __restrict__ rowA = per_v_e + (size_t)(row0 + l15) * Ff; // K 0..127
  const float* __restrict__ rowB = v_in + (size_t)(row0 + l15) * Ff;    // K 128..255

  v8f acc[8] = {};
  for (int k = 0; k < 2 * Ff; k += 4) {
    const int kb = k + 2 * half;                 // (kb,kb+1) never straddles 128
    const float* src = (kb < Ff) ? (rowA + kb) : (rowB + (kb - Ff));
    v2f a = *(const v2f*)src;
    const float* __restrict__ w0 = W_v + (size_t)kb * Ff;
#pragma unroll
    for (int t = 0; t < 8; ++t) {
      const int g = t * 16 + l15;
      v2f bf;
      bf.x = w0[g];
      bf.y = w0[Ff + g];
      acc[t] = wmma_f32_k4(a, bf, acc[t]);
    }
  }
#pragma unroll
  for (int t = 0; t < 8; ++t) {
    const int g = t * 16 + l15;
    const float bias = b_v[g];
#pragma unroll
    for (int r = 0; r < 8; ++r) {
      const int mrow = r + half * 8;
      v_out[(size_t)(row0 + mrow) * Ff + g] = fmaxf(acc[t][r] + bias, 0.0f);
    }
  }
}

// ---------------------------------------------------------------------------
extern "C" void kernel_launch(void* const* d_in, const int* in_sizes, int n_in,
                              void* d_out, int out_size, void* d_ws, size_t ws_size,
                              hipStream_t stream) {
  (void)in_sizes; (void)n_in; (void)out_size; (void)ws_size;
  const float* v_in = (const float*)d_in[0];
  const float* e_in = (const float*)d_in[1];
  const float* W_ev = (const float*)d_in[2];
  const float* b_ev = (const float*)d_in[3];
  const float* W_e  = (const float*)d_in[4];
  const float* b_e  = (const float*)d_in[5];
  const float* W_v  = (const float*)d_in[6];
  const float* b_v  = (const float*)d_in[7];

  float* out   = (float*)d_out;
  float* v_out = out;                          // [32,128,128]
  float* e_out = out + (size_t)ROWS * Ff;      // [32,128,128,128]

  float* e_v      = (float*)d_ws;                            // 4096*128 = 2MB
  float* per_v_e  = e_v + (size_t)ROWS * Ff;                 // 2MB
  float* partials = per_v_e + (size_t)ROWS * Ff;             // 32*8*128*128 = 16MB

  ev_kernel<<<64, 128, 0, stream>>>(v_in, W_ev, b_ev, e_v);
  edge_kernel<<<dim3(8, 8, 32), 256, 0, stream>>>(e_in, W_e, b_e, e_v, e_out, partials);
  reduce_kernel<<<(ROWS * Ff + 255) / 256, 256, 0, stream>>>(partials, per_v_e);
  vert_kernel<<<64, 128, 0, stream>>>(per_v_e, v_in, W_v, b_v, v_out);
}